// KeypointDetction_55336358641995
// MI455X (gfx1250) — compile-verified
//
#include <hip/hip_runtime.h>
#include <hip/hip_bf16.h>
#include <float.h>

// ---------------- problem constants (from reference) ----------------
#define IMB   16
#define IMH   1024
#define IMW   1024
#define IMHW  (IMH * IMW)
#define TOPK  512
#define QUALITY 0.3f
#define NMSWIN  15          // 2*MIN_DIST+1
#define NMSR    7
#define CAP     8192        // candidate capacity per image (>= realistic NMS survivors)
#define SORTN   8192        // power of two == CAP

typedef float v2f __attribute__((ext_vector_type(2)));
typedef float v8f __attribute__((ext_vector_type(8)));

// ---------------- workspace layout ----------------
// [0]                : unsigned long long cand[IMB*CAP]   (1 MiB)
// [CAND_BYTES]       : unsigned int maxBits[16]           (64 B)
// [CAND_BYTES+64]    : unsigned int cnt[16]               (64 B)
#define CAND_BYTES ((size_t)IMB * CAP * 8)

// ======================================================================
// Kernel 1: fused quantize -> Sobel -> structure tensor (7x7 box) ->
//           Shi-Tomasi response. Vertical 7-tap box via WMMA f32 16x16x4.
// One wave per 16x16 output tile; 4 waves per block (block tile 64x16).
// ======================================================================

// per-wave LDS layout (floats)
#define K1_IMG_OFF   0            // 24 x 24, stride 25  -> 600
#define K1_PA_OFF    600          // 22 x 22, stride 23  -> 506
#define K1_PB_OFF    1106
#define K1_PC_OFF    1612
#define K1_MA_OFF    2118         // 24 x 16, stride 17  -> 408 (rows 22..23 zeroed)
#define K1_MB_OFF    2526
#define K1_MC_OFF    2934
#define K1_WAVE_FLTS 3342

__device__ __forceinline__ float bandv(unsigned r, unsigned k) {
    // Band(16x24): out-row r sums mid rows r..r+6 (image rows +/-3)
    return (k >= r && k <= r + 6u) ? 1.0f : 0.0f;
}

__global__ __launch_bounds__(128) void k1_response(const float* __restrict__ img,
                                                   float* __restrict__ resp,
                                                   unsigned int* __restrict__ maxBits) {
    __shared__ float smem[4 * K1_WAVE_FLTS];

    const unsigned tid  = threadIdx.x;
    const unsigned lane = tid & 31u;
    const unsigned wv   = tid >> 5;

    float* S  = smem + wv * K1_WAVE_FLTS;
    float* sImg = S + K1_IMG_OFF;
    float* sPA  = S + K1_PA_OFF;
    float* sPB  = S + K1_PB_OFF;
    float* sPC  = S + K1_PC_OFF;
    float* sMA  = S + K1_MA_OFF;
    float* sMB  = S + K1_MB_OFF;
    float* sMC  = S + K1_MC_OFF;

    const int b     = blockIdx.z;
    const int tileX = (blockIdx.x * 4 + wv) * 16;
    const int tileY = blockIdx.y * 16;
    const float* imB = img + (size_t)b * IMHW;

    // ---- load 24x24 quantized tile (zero pad outside image) ----
    for (unsigned i = lane; i < 24u * 24u; i += 32u) {
        const int r = (int)(i / 24u), c = (int)(i % 24u);
        const int gy = tileY - 4 + r, gx = tileX - 4 + c;
        float v = 0.0f;
        if (gy >= 0 && gy < IMH && gx >= 0 && gx < IMW)
            v = floorf(imB[(size_t)gy * IMW + gx] * 255.0f);
        sImg[r * 25 + c] = v;
    }
    __syncthreads();

    // ---- Sobel + structure-tensor products at tile +/-3 (22x22) ----
    for (unsigned i = lane; i < 22u * 22u; i += 32u) {
        const int k = (int)(i / 22u), j = (int)(i % 22u);
        const int py = tileY - 3 + k, px = tileX - 3 + j;
        float pa = 0.0f, pb = 0.0f, pc = 0.0f;
        if (py >= 0 && py < IMH && px >= 0 && px < IMW) {
            const float* T = sImg + (k + 1) * 25 + (j + 1);
            const float tl = T[-25 - 1], tc = T[-25], tr = T[-25 + 1];
            const float ml = T[-1],                   mr = T[+1];
            const float bl = T[+25 - 1], bc = T[+25], br = T[+25 + 1];
            const float ix = (tr + 2.0f * mr + br) - (tl + 2.0f * ml + bl);
            const float iy = (bl + 2.0f * bc + br) - (tl + 2.0f * tc + tr);
            pa = ix * ix; pb = ix * iy; pc = iy * iy;
        }
        sPA[k * 23 + j] = pa;
        sPB[k * 23 + j] = pb;
        sPC[k * 23 + j] = pc;
    }
    __syncthreads();

    // ---- horizontal 7-tap box (VALU) -> mid (24x16, rows 22..23 zeroed) ----
    for (unsigned i = lane; i < 24u * 16u; i += 32u) {      // 12 uniform iters/lane
        const unsigned k = i >> 4, c = i & 15u;
        float ma = 0.0f, mb = 0.0f, mc = 0.0f;
        if (k < 22u) {
            const float* pa = sPA + k * 23 + c;
            const float* pb = sPB + k * 23 + c;
            const float* pc = sPC + k * 23 + c;
            #pragma unroll
            for (int t = 0; t < 7; ++t) { ma += pa[t]; mb += pb[t]; mc += pc[t]; }
        }
        sMA[k * 17 + c] = ma;
        sMB[k * 17 + c] = mb;
        sMC[k * 17 + c] = mc;
    }
    __syncthreads();

    // ---- vertical 7-tap box via WMMA: D(16x16) = Band(16x24) x Mid(24x16) ----
    v8f accA = {}, accB = {}, accC = {};
    const unsigned half = lane >> 4;            // 0 or 1
    const unsigned rc   = lane & 15u;           // A: row, B: col
    #pragma unroll
    for (unsigned kc = 0; kc < 6; ++kc) {
        const unsigned kb = kc * 4u + (half ? 2u : 0u);
        v2f a; a.x = bandv(rc, kb); a.y = bandv(rc, kb + 1u);
        v2f bA; bA.x = sMA[kb * 17 + rc]; bA.y = sMA[(kb + 1) * 17 + rc];
        v2f bB; bB.x = sMB[kb * 17 + rc]; bB.y = sMB[(kb + 1) * 17 + rc];
        v2f bC; bC.x = sMC[kb * 17 + rc]; bC.y = sMC[(kb + 1) * 17 + rc];
        accA = __builtin_amdgcn_wmma_f32_16x16x4_f32(false, a, false, bA, (short)0, accA, false, false);
        accB = __builtin_amdgcn_wmma_f32_16x16x4_f32(false, a, false, bB, (short)0, accB, false, false);
        accC = __builtin_amdgcn_wmma_f32_16x16x4_f32(false, a, false, bC, (short)0, accC, false, false);
    }

    // ---- Shi-Tomasi response + store + per-image max ----
    float m = 0.0f;
    float* respB = resp + (size_t)b * IMHW;
    const int gx = tileX + (int)rc;
    #pragma unroll
    for (int r = 0; r < 8; ++r) {
        const int row = r + (half ? 8 : 0);
        const float a = accA[r], bb = accB[r], c = accC[r];
        const float v = 0.5f * ((a + c) - sqrtf((a - c) * (a - c) + 4.0f * bb * bb));
        respB[(size_t)(tileY + row) * IMW + gx] = v;
        m = fmaxf(m, v);
    }
    #pragma unroll
    for (int off = 16; off >= 1; off >>= 1)
        m = fmaxf(m, __shfl_xor(m, off, 32));
    if (lane == 0)
        atomicMax(&maxBits[b], __float_as_uint(fmaxf(m, 0.0f)));
}

// ======================================================================
// Kernel 2: 15x15 separable NMS + quality threshold -> candidate list.
// Block = 256 threads, tile 32x32, halo 7.
// ======================================================================
__global__ __launch_bounds__(256) void k2_nms(const float* __restrict__ resp,
                                              const unsigned int* __restrict__ maxBits,
                                              unsigned int* __restrict__ cnt,
                                              unsigned long long* __restrict__ cand) {
    __shared__ float sR[46 * 47];
    __shared__ float sH[46 * 33];

    const int b     = blockIdx.z;
    const int tileX = blockIdx.x * 32;
    const int tileY = blockIdx.y * 32;
    const float* respB = resp + (size_t)b * IMHW;
    const float thresh = QUALITY * __uint_as_float(maxBits[b]);

    for (unsigned i = threadIdx.x; i < 46u * 46u; i += 256u) {
        const int r = (int)(i / 46u), c = (int)(i % 46u);
        const int gy = tileY - NMSR + r, gx = tileX - NMSR + c;
        float v = -FLT_MAX;
        if (gy >= 0 && gy < IMH && gx >= 0 && gx < IMW)
            v = respB[(size_t)gy * IMW + gx];
        sR[r * 47 + c] = v;
    }
    __syncthreads();

    for (unsigned i = threadIdx.x; i < 46u * 32u; i += 256u) {
        const unsigned r = i >> 5, c = i & 31u;
        const float* p = sR + r * 47 + c;
        float m = p[0];
        #pragma unroll
        for (int t = 1; t < NMSWIN; ++t) m = fmaxf(m, p[t]);
        sH[r * 33 + c] = m;
    }
    __syncthreads();

    for (unsigned i = threadIdx.x; i < 32u * 32u; i += 256u) {
        const unsigned r = i >> 5, c = i & 31u;
        const float* p = sH + r * 33 + c;
        float lmax = p[0];
        #pragma unroll
        for (int t = 1; t < NMSWIN; ++t) lmax = fmaxf(lmax, p[t * 33]);
        const float v = sR[(r + NMSR) * 47 + (c + NMSR)];
        if (v >= lmax && v >= thresh) {
            const unsigned gidx = (unsigned)(tileY + (int)r) * IMW + (unsigned)(tileX + (int)c);
            const unsigned slot = atomicAdd(&cnt[b], 1u);
            if (slot < CAP) {
                // key: desc by value bits, ties -> ascending index
                const unsigned long long key =
                    ((unsigned long long)__float_as_uint(v) << 32) |
                    (unsigned long long)(0xFFFFFFFFu - gidx);
                cand[(size_t)b * CAP + slot] = key;
            }
        }
    }
}

// ======================================================================
// Kernel 3: per image, bitonic sort 8192 keys desc in LDS, emit top-512
// coords (sorted) and scatter 1.0 into the (pre-zeroed) mask.
// ======================================================================
__global__ __launch_bounds__(1024) void k3_topk(const unsigned long long* __restrict__ cand,
                                                const unsigned int* __restrict__ cnt,
                                                float* __restrict__ mask,
                                                float* __restrict__ coords) {
    __shared__ unsigned long long key[SORTN];   // 64 KiB

    const int b = blockIdx.x;
    unsigned n = cnt[b];
    if (n > CAP) n = CAP;

    for (unsigned i = threadIdx.x; i < SORTN; i += 1024u)
        key[i] = (i < n) ? cand[(size_t)b * CAP + i] : 0ull;
    __syncthreads();

    // bitonic sort, descending
    for (unsigned k = 2; k <= SORTN; k <<= 1) {
        for (unsigned j = k >> 1; j >= 1; j >>= 1) {
            for (unsigned i = threadIdx.x; i < SORTN; i += 1024u) {
                const unsigned l = i ^ j;
                if (l > i) {
                    const unsigned long long a = key[i], c = key[l];
                    const bool desc = ((i & k) == 0u);
                    if (desc ? (a < c) : (a > c)) { key[i] = c; key[l] = a; }
                }
            }
            __syncthreads();
        }
    }

    if (threadIdx.x < TOPK) {
        const unsigned s = threadIdx.x;
        const bool valid = (s < n);
        const unsigned long long kk = key[s];
        const unsigned gidx = 0xFFFFFFFFu - (unsigned)(kk & 0xFFFFFFFFull);
        const unsigned y = gidx >> 10;       // W = 1024
        const unsigned x = gidx & 1023u;
        float* co = coords + ((size_t)b * TOPK + s) * 2;
        co[0] = valid ? (float)y : -1.0f;
        co[1] = valid ? (float)x : -1.0f;
        if (valid)
            mask[(size_t)b * IMHW + gidx] = 1.0f;
    }
}

// ======================================================================
extern "C" void kernel_launch(void* const* d_in, const int* in_sizes, int n_in,
                              void* d_out, int out_size, void* d_ws, size_t ws_size,
                              hipStream_t stream) {
    (void)in_sizes; (void)n_in; (void)out_size; (void)ws_size;

    const float* img = (const float*)d_in[0];
    float* out = (float*)d_out;
    float* mask   = out;                         // [16,1,1024,1024]
    float* coords = out + (size_t)IMB * IMHW;    // [16,512,2]

    unsigned long long* cand = (unsigned long long*)d_ws;
    unsigned int* maxBits = (unsigned int*)((char*)d_ws + CAND_BYTES);
    unsigned int* cnt     = maxBits + 16;

    // zero per-image max + counters
    hipMemsetAsync((char*)d_ws + CAND_BYTES, 0, 128, stream);

    // response into the d_out mask region (reused as scratch; rewritten later)
    dim3 g1(IMW / 64, IMH / 16, IMB);
    k1_response<<<g1, 128, 0, stream>>>(img, mask, maxBits);

    dim3 g2(IMW / 32, IMH / 32, IMB);
    k2_nms<<<g2, 256, 0, stream>>>(mask, maxBits, cnt, cand);

    // resp consumed; zero the mask before scatter
    hipMemsetAsync(mask, 0, (size_t)IMB * IMHW * sizeof(float), stream);

    k3_topk<<<IMB, 1024, 0, stream>>>(cand, cnt, mask, coords);
}